// HierarchicalPoincareRest_58763742544823
// MI455X (gfx1250) — compile-verified
//
#include <hip/hip_runtime.h>
#include <math.h>

typedef __attribute__((ext_vector_type(2))) float v2f;
typedef __attribute__((ext_vector_type(4))) float v4f;
typedef __attribute__((ext_vector_type(8))) float v8f;
typedef __attribute__((__vector_size__(4 * sizeof(int)))) int v4i;

#define C_CURV   1.5f
#define SQRT_C   1.22474487139158905f
#define INV_TEMP (1.0f / 0.07f)

#if __has_builtin(__builtin_amdgcn_global_load_async_to_lds_b128) && \
    __has_builtin(__builtin_amdgcn_s_wait_asynccnt)
#define USE_ASYNC 1
#define ASYNC_WAIT(n) __builtin_amdgcn_s_wait_asynccnt(n)
#else
#define USE_ASYNC 0
#define ASYNC_WAIT(n) ((void)0)
#endif

__device__ __forceinline__ float gelu_exact(float x) {
    // jax.nn.gelu(approximate=False): x * 0.5 * (1 + erf(x / sqrt(2)))
    return 0.5f * x * (1.0f + erff(x * 0.70710678118654752440f));
}

__device__ __forceinline__ void lds_copy16(float* dst_lds, const float* src_glb) {
#if USE_ASYNC
    __builtin_amdgcn_global_load_async_to_lds_b128(
        (v4i*)src_glb, (v4i*)dst_lds, 0, 0);
#else
    *(v4f*)dst_lds = *(const v4f*)src_glb;
#endif
}

// ---------------------------------------------------------------------------
// C[64 x N] = act(A[64 x K] @ W[K x N] + bias[N]) per level (blockIdx.z).
// Block (256 thr, 8 waves) computes a 64x64 tile of C.  K-chunks of 16 are
// double-buffered through LDS via async b128 global->LDS copies; fragments
// are read from LDS and fed to V_WMMA_F32_16X16X4_F32.
//   wave w: m-tile = w&3 (16 rows), n-half = w>>2 (32 cols -> 2 accum tiles)
// ---------------------------------------------------------------------------
#define KB    16
#define A_PAD 18   // even (keeps 8B alignment of ds b64) and stride 18 -> no bank conflicts

__global__ __launch_bounds__(256)
void gemm_wmma_f32(const float* __restrict__ A, long a_lvl_stride,
                   const float* __restrict__ W, long w_lvl_stride,
                   const float* __restrict__ bias, long b_lvl_stride,
                   float* __restrict__ C, long c_lvl_stride,
                   int K, int N, int do_gelu)
{
    __shared__ float As[2][64 * A_PAD];
    __shared__ float Ws[2][KB * 64];

    const int z = blockIdx.z;
    const float* Ab = A    + (size_t)z * a_lvl_stride;
    const float* Wb = W    + (size_t)z * w_lvl_stride;
    const float* bb = bias + (size_t)z * b_lvl_stride;
    float*       Cb = C    + (size_t)z * c_lvl_stride;

    const int tid  = threadIdx.x;
    const int wid  = tid >> 5;
    const int lane = tid & 31;
    const int half = lane >> 4;
    const int mr   = lane & 15;

    const int n0base = blockIdx.x * 64;

    // per-thread copy slots
    const int aRow = tid >> 2, aCol = (tid & 3) * 4;       // A: 64 rows x 16 cols
    const int wRow = tid >> 4, wCol = (tid & 15) * 4;      // W: 16 rows x 64 cols
    const float* aSrc = Ab + (size_t)aRow * K + aCol;
    const float* wSrc = Wb + (size_t)wRow * N + n0base + wCol;

    // per-wave compute slots
    const int m0 = (wid & 3) * 16;
    const int nb = (wid >> 2) * 32;

    v8f acc[2];
    #pragma unroll
    for (int t = 0; t < 2; ++t)
        #pragma unroll
        for (int e = 0; e < 8; ++e) acc[t][e] = 0.0f;

    const int nChunks = K / KB;

    // prologue: stage chunk 0 into buffer 0
    lds_copy16(&As[0][aRow * A_PAD + aCol], aSrc);
    lds_copy16(&Ws[0][wRow * 64 + wCol], wSrc);
    aSrc += KB;
    wSrc += (size_t)KB * N;

    for (int kc = 0; kc < nChunks; ++kc) {
        const int cur = kc & 1;
        if (kc + 1 < nChunks) {
            lds_copy16(&As[cur ^ 1][aRow * A_PAD + aCol], aSrc);
            lds_copy16(&Ws[cur ^ 1][wRow * 64 + wCol], wSrc);
            aSrc += KB;
            wSrc += (size_t)KB * N;
            ASYNC_WAIT(2);   // current chunk landed; next chunk stays in flight
        } else {
            ASYNC_WAIT(0);
        }
        __syncthreads();

        const float* aL = &As[cur][(m0 + mr) * A_PAD];
        const float* wL = &Ws[cur][nb + mr];
        #pragma unroll
        for (int kk = 0; kk < 4; ++kk) {
            const int ka = kk * 4 + 2 * half;
            const v2f a = *(const v2f*)(aL + ka);
            #pragma unroll
            for (int t = 0; t < 2; ++t) {
                v2f b;
                b.x = wL[ka * 64 + t * 16];
                b.y = wL[(ka + 1) * 64 + t * 16];
                acc[t] = __builtin_amdgcn_wmma_f32_16x16x4_f32(
                    false, a, false, b, (short)0, acc[t], false, false);
            }
        }
        __syncthreads();   // everyone done reading before buffer reuse
    }

    #pragma unroll
    for (int t = 0; t < 2; ++t) {
        const int n  = n0base + nb + t * 16 + mr;
        const float bv = bb[n];
        #pragma unroll
        for (int r = 0; r < 8; ++r) {
            const int m = m0 + r + 8 * half;   // C VGPR r: lanes0-15 M=r, lanes16-31 M=r+8
            float v = acc[t][r] + bv;
            if (do_gelu) v = gelu_exact(v);
            Cb[(size_t)m * N + n] = v;
        }
    }
}

// ---------------------------------------------------------------------------
// expmap0 over rows of 128, also emits f2 = ||emb||^2 (needed by hyp dist)
// ---------------------------------------------------------------------------
__global__ __launch_bounds__(128)
void expmap_f2_kernel(const float* __restrict__ feats,
                      float* __restrict__ emb, float* __restrict__ f2, int rows)
{
    const int wid = threadIdx.x >> 5, lane = threadIdx.x & 31;
    const int row = blockIdx.x * (blockDim.x >> 5) + wid;
    if (row >= rows) return;
    const float* u = feats + (size_t)row * 128;
    float v[4], s = 0.0f;
    #pragma unroll
    for (int i = 0; i < 4; ++i) { v[i] = u[lane + 32 * i]; s += v[i] * v[i]; }
    #pragma unroll
    for (int off = 16; off; off >>= 1) s += __shfl_xor(s, off, 32);
    const float n2    = fmaxf(s, 1e-15f);
    const float nn    = sqrtf(n2);
    const float scale = tanhf(SQRT_C * nn) / (SQRT_C * nn);
    float* e = emb + (size_t)row * 128;
    #pragma unroll
    for (int i = 0; i < 4; ++i) e[lane + 32 * i] = v[i] * scale;
    if (lane == 0) f2[row] = scale * scale * n2;
}

// ---------------------------------------------------------------------------
// p2[i] = ||P[i]||^2  (one wave per 128-wide row)
// ---------------------------------------------------------------------------
__global__ __launch_bounds__(128)
void rownorm2_kernel(const float* __restrict__ P, float* __restrict__ p2, int rows)
{
    const int wid = threadIdx.x >> 5, lane = threadIdx.x & 31;
    const int row = blockIdx.x * (blockDim.x >> 5) + wid;
    if (row >= rows) return;
    const float* u = P + (size_t)row * 128;
    float s = 0.0f;
    #pragma unroll
    for (int i = 0; i < 4; ++i) { float x = u[lane + 32 * i]; s += x * x; }
    #pragma unroll
    for (int off = 16; off; off >>= 1) s += __shfl_xor(s, off, 32);
    if (lane == 0) p2[row] = s;
}

// ---------------------------------------------------------------------------
// logits[b, j] = -hyp_dist(emb[b], proto[j]) / TEMP
// dot = emb @ proto^T via WMMA (K=128); fragments preloaded in batches of 16
// k-steps so the 16 WMMAs issue back-to-back after a single wait; then the
// closed-form ||ma||^2 = (cx^2*p2 - 2*cx*cy*dot + cy^2*f2)/denom^2 epilogue.
// ---------------------------------------------------------------------------
__global__ __launch_bounds__(256)
void hypdist_wmma(const float* __restrict__ emb, const float* __restrict__ proto,
                  const float* __restrict__ p2, const float* __restrict__ f2,
                  float* __restrict__ out, int np)
{
    const int wid  = threadIdx.x >> 5;
    const int lane = threadIdx.x & 31;
    const int half = lane >> 4;
    const int mr   = lane & 15;

    const int jTiles = (np + 15) >> 4;
    const int tiles  = jTiles * 4;
    const int flat   = blockIdx.x * 8 + wid;
    if (flat >= tiles) return;
    const int m0 = (flat & 3) << 4;
    const int j  = ((flat >> 2) << 4) + mr;
    const bool jvalid = (j < np);
    const float msk = jvalid ? 1.0f : 0.0f;

    const float* arow = emb   + (size_t)(m0 + mr) * 128;
    const float* prow = proto + (size_t)(jvalid ? j : 0) * 128;

    v8f acc;
    #pragma unroll
    for (int e = 0; e < 8; ++e) acc[e] = 0.0f;

    #pragma unroll
    for (int h2 = 0; h2 < 2; ++h2) {
        v2f af[16], bf[16];
        #pragma unroll
        for (int kk = 0; kk < 16; ++kk) {
            const int ka = h2 * 64 + kk * 4 + 2 * half;
            af[kk] = *(const v2f*)(arow + ka);
            v2f b  = *(const v2f*)(prow + ka);
            bf[kk].x = b.x * msk;
            bf[kk].y = b.y * msk;
        }
        #pragma unroll
        for (int kk = 0; kk < 16; ++kk)
            acc = __builtin_amdgcn_wmma_f32_16x16x4_f32(
                false, af[kk], false, bf[kk], (short)0, acc, false, false);
    }

    const float p2n = jvalid ? p2[j] : 0.0f;
    const float cy  = 1.0f - C_CURV * p2n;
    #pragma unroll
    for (int r = 0; r < 8; ++r) {
        const int m = m0 + r + 8 * half;
        const float dot   = acc[r];
        const float f2m   = f2[m];
        const float denom = 1.0f - 2.0f * C_CURV * dot + C_CURV * C_CURV * f2m * p2n;
        const float cx    = 1.0f - 2.0f * C_CURV * dot + C_CURV * f2m;
        float n2 = (cx * cx * p2n - 2.0f * cx * cy * dot + cy * cy * f2m) / (denom * denom);
        n2 = fmaxf(n2, 1e-15f);
        const float arg  = fminf(fmaxf(SQRT_C * sqrtf(n2), 0.0f), 1.0f - 1e-7f);
        const float dist = (2.0f / SQRT_C) * atanhf(arg);
        if (jvalid) out[(size_t)m * np + j] = -dist * INV_TEMP;
    }
}

// ---------------------------------------------------------------------------
// logits0[b, c] = sum_j (parent[j]==c) * logits1[b, j]  (deterministic gather)
// ---------------------------------------------------------------------------
__global__ __launch_bounds__(256)
void parent_gather_kernel(const float* __restrict__ logits1,
                          const int* __restrict__ parent, float* __restrict__ out0)
{
    const int tid = blockIdx.x * blockDim.x + threadIdx.x;
    if (tid >= 64 * 100) return;
    const int b = tid / 100, c = tid % 100;
    const float* row = logits1 + (size_t)b * 1000;
    float acc = 0.0f;
    for (int jj = 0; jj < 1000; ++jj)
        acc += (parent[jj] == c) ? row[jj] : 0.0f;
    out0[tid] = acc;
}

// ---------------------------------------------------------------------------
extern "C" void kernel_launch(void* const* d_in, const int* in_sizes, int n_in,
                              void* d_out, int out_size, void* d_ws, size_t ws_size,
                              hipStream_t stream)
{
    const float* x     = (const float*)d_in[0];   // 64 x 2048
    const float* W1    = (const float*)d_in[1];   // 3 x 2048 x 4096
    const float* b1    = (const float*)d_in[2];   // 3 x 4096
    const float* W2    = (const float*)d_in[3];   // 3 x 4096 x 4096
    const float* b2    = (const float*)d_in[4];   // 3 x 4096
    const float* W3    = (const float*)d_in[5];   // 3 x 4096 x 128
    const float* b3    = (const float*)d_in[6];   // 3 x 128
    const float* proto = (const float*)d_in[7];   // 11100 x 128
    const int*   par   = (const int*)d_in[8];     // 1000

    float* out   = (float*)d_out;
    float* out0  = out;                    // 64 x 100
    float* out1  = out + 6400;             // 64 x 1000
    float* out2  = out + 6400 + 64000;     // 64 x 10000

    // workspace layout (floats); only levels 1 and 2 are live
    float* ws    = (float*)d_ws;
    float* H1    = ws;                     // 2 x 64 x 4096
    float* H2    = H1 + 2 * 64 * 4096;     // 2 x 64 x 4096
    float* feats = H2 + 2 * 64 * 4096;     // 2 x 64 x 128
    float* embp  = feats + 2 * 64 * 128;   // 2 x 64 x 128
    float* f2w   = embp + 2 * 64 * 128;    // 128
    float* p2w   = f2w + 128;              // 11000

    // --- layer 1: gelu(x @ W1[l] + b1[l]), levels 1..2 ---
    {
        dim3 g(4096 / 64, 1, 2);
        gemm_wmma_f32<<<g, 256, 0, stream>>>(
            x, 0L,
            W1 + (size_t)1 * 2048 * 4096, (long)2048 * 4096,
            b1 + 4096, 4096L,
            H1, (long)64 * 4096,
            2048, 4096, 1);
    }
    // --- layer 2: gelu(H1 @ W2[l] + b2[l]) ---
    {
        dim3 g(4096 / 64, 1, 2);
        gemm_wmma_f32<<<g, 256, 0, stream>>>(
            H1, (long)64 * 4096,
            W2 + (size_t)1 * 4096 * 4096, (long)4096 * 4096,
            b2 + 4096, 4096L,
            H2, (long)64 * 4096,
            4096, 4096, 1);
    }
    // --- layer 3: H2 @ W3[l] + b3[l] ---
    {
        dim3 g(128 / 64, 1, 2);
        gemm_wmma_f32<<<g, 256, 0, stream>>>(
            H2, (long)64 * 4096,
            W3 + (size_t)1 * 4096 * 128, (long)4096 * 128,
            b3 + 128, 128L,
            feats, (long)64 * 128,
            4096, 128, 0);
    }
    // --- expmap0 + f2 (128 rows = 2 levels x 64) ---
    expmap_f2_kernel<<<32, 128, 0, stream>>>(feats, embp, f2w, 128);
    // --- prototype norms for rows [100, 11100) ---
    rownorm2_kernel<<<(11000 + 3) / 4, 128, 0, stream>>>(proto + (size_t)100 * 128, p2w, 11000);
    // --- distances ---
    {
        int tiles = ((1000 + 15) / 16) * 4;
        hypdist_wmma<<<(tiles + 7) / 8, 256, 0, stream>>>(
            embp, proto + (size_t)100 * 128, p2w, f2w, out1, 1000);
    }
    {
        int tiles = ((10000 + 15) / 16) * 4;
        hypdist_wmma<<<(tiles + 7) / 8, 256, 0, stream>>>(
            embp + 64 * 128, proto + (size_t)1100 * 128, p2w + 1000, f2w + 64, out2, 10000);
    }
    // --- logits0 via deterministic parent gather ---
    parent_gather_kernel<<<(6400 + 255) / 256, 256, 0, stream>>>(out1, par, out0);
}